// InfoNCELoss_29755533426796
// MI455X (gfx1250) — compile-verified
//
#include <hip/hip_runtime.h>

// InfoNCE loss, fully fused: 4 Gram passes (11, 12, 21=12^T, 22) with
// V_WMMA_F32_16X16X4_F32 + online logsumexp. No N x N materialization.
// B=16, N=2048, D=256, fp32 (matches reference setup_inputs()).
//
// Block = 128 threads = 4 waves. Each block owns one (batch, 16-row tile);
// each wave handles a 32-column-tile chunk of the j loop (occupancy: 8192
// waves). Per-chunk LSE partials merge through LDS at block end.
// K loop is software-pipelined: next K-block operands load while current
// K-block's 4 WMMAs execute.

typedef float v2f __attribute__((ext_vector_type(2)));
typedef float v8f __attribute__((ext_vector_type(8)));

#define BB 16
#define NN 2048
#define DD 256
#define NTILE (NN / 16)          // 128 column tiles
#define WAVES 4
#define CHUNK (NTILE / WAVES)    // 32 j-tiles per wave
#define INV_TEMP 10.0f
#define NEG_BIG (-1e30f)

__device__ __forceinline__ void lse_update(float& m, float& s, float x) {
    float mn = fmaxf(m, x);
    s = s * __expf(m - mn) + __expf(x - mn);
    m = mn;
}

__device__ __forceinline__ void lse_merge(float& m, float& s, float om, float os) {
    float mn = fmaxf(m, om);
    s = s * __expf(m - mn) + os * __expf(om - mn);
    m = mn;
}

#define GRAM4()                                                                   \
    acc11 = __builtin_amdgcn_wmma_f32_16x16x4_f32(false, a1c, false, b1c,         \
                                                  (short)0, acc11, false, false); \
    acc12 = __builtin_amdgcn_wmma_f32_16x16x4_f32(false, a1c, false, b2c,         \
                                                  (short)0, acc12, false, false); \
    acc21 = __builtin_amdgcn_wmma_f32_16x16x4_f32(false, a2c, false, b1c,         \
                                                  (short)0, acc21, false, false); \
    acc22 = __builtin_amdgcn_wmma_f32_16x16x4_f32(false, a2c, false, b2c,         \
                                                  (short)0, acc22, false, false)

__global__ void __launch_bounds__(128)
infonce_fused_kernel(const float* __restrict__ X1, const float* __restrict__ X2,
                     float* __restrict__ out) {
    const int task = blockIdx.x;            // 0 .. B*(N/16)-1
    const int b    = task / NTILE;
    const int i    = task % NTILE;          // row tile index
    const int tid  = threadIdx.x;
    const int wave = tid >> 5;              // 0..3: j-chunk owner
    const int lane = tid & 31;              // wave32 lane
    const int col  = lane & 15;             // N index within 16-tile (B/C/D layout)
    const int half = lane >> 4;             // K-half selector for A/B operands

    __shared__ float lm1[WAVES][16], ls1[WAVES][16];
    __shared__ float lm2[WAVES][16], ls2[WAVES][16];
    __shared__ float lpos[16];

    const size_t batchOff = (size_t)b * NN * DD;
    // A operand: lane holds row (i*16 + lane%16), float2 at K = 4*kb + 2*half.
    // B operand for tile j uses the *same* per-lane addressing on X[j] rows
    // (f32 16x16x4 A and B VGPR layouts are index-identical -> Gram is free).
    const float* a1p = X1 + batchOff + (size_t)(i * 16 + col) * DD + 2 * half;
    const float* a2p = X2 + batchOff + (size_t)(i * 16 + col) * DD + 2 * half;

    // Running LSE state: 8 rows per lane (C/D layout: row = v + 8*half).
    // set1: rows of [logits_11 | logits_12]  (loss_1 denominator)
    // set2: rows of [logits_21 | logits_22]  (loss_2 denominator; 21 = 12^T)
    float m1[8], s1[8], m2[8], s2[8], pos[8];
#pragma unroll
    for (int v = 0; v < 8; ++v) {
        m1[v] = NEG_BIG; s1[v] = 0.f;
        m2[v] = NEG_BIG; s2[v] = 0.f;
        pos[v] = 0.f;
    }

    const int j0 = wave * CHUNK;
    for (int j = j0; j < j0 + CHUNK; ++j) {
        const float* b1p = X1 + batchOff + (size_t)(j * 16 + col) * DD + 2 * half;
        const float* b2p = X2 + batchOff + (size_t)(j * 16 + col) * DD + 2 * half;

        v8f acc11 = {}; v8f acc12 = {}; v8f acc21 = {}; v8f acc22 = {};

        // Software-pipelined K loop: load next K-block while WMMAing current.
        v2f a1c = *(const v2f*)(a1p);
        v2f a2c = *(const v2f*)(a2p);
        v2f b1c = *(const v2f*)(b1p);
        v2f b2c = *(const v2f*)(b2p);
#pragma unroll 7
        for (int kb = 0; kb < DD / 4 - 1; ++kb) {
            v2f a1n = *(const v2f*)(a1p + 4 * (kb + 1));
            v2f a2n = *(const v2f*)(a2p + 4 * (kb + 1));
            v2f b1n = *(const v2f*)(b1p + 4 * (kb + 1));
            v2f b2n = *(const v2f*)(b2p + 4 * (kb + 1));
            GRAM4();
            a1c = a1n; a2c = a2n; b1c = b1n; b2c = b2n;
        }
        GRAM4();   // epilogue: last K-block

        // Fold this 16-column slab into the running row-LSEs.
#pragma unroll
        for (int v = 0; v < 8; ++v) {
            int  row  = v + 8 * half;                  // tile-local row (M)
            bool diag = (i == j) && (row == col);      // global diagonal element
            float x11 = acc11[v] * INV_TEMP;
            float x12 = acc12[v] * INV_TEMP;
            float x21 = acc21[v] * INV_TEMP;
            float x22 = acc22[v] * INV_TEMP;
            if (diag) { pos[v] = x12; x11 = NEG_BIG; x22 = NEG_BIG; }
            lse_update(m1[v], s1[v], x11);
            lse_update(m1[v], s1[v], x12);
            lse_update(m2[v], s2[v], x21);
            lse_update(m2[v], s2[v], x22);
        }
    }

    // Merge partial LSEs across the 16 lanes of each half-group
    // (row r's 16 column-partials live in lanes {16*half + 0..15}).
#pragma unroll
    for (int mask = 1; mask < 16; mask <<= 1) {
#pragma unroll
        for (int v = 0; v < 8; ++v) {
            float om = __shfl_xor(m1[v], mask, 32);
            float os = __shfl_xor(s1[v], mask, 32);
            lse_merge(m1[v], s1[v], om, os);
            om = __shfl_xor(m2[v], mask, 32);
            os = __shfl_xor(s2[v], mask, 32);
            lse_merge(m2[v], s2[v], om, os);
        }
    }

    // Owner lane of each row (lane%16 == row) exports this wave's chunk state.
    {
        int v = col - 8 * half;             // owning v, valid iff 0 <= v < 8
        if (v >= 0 && v < 8) {
            lm1[wave][col] = m1[v]; ls1[wave][col] = s1[v];
            lm2[wave][col] = m2[v]; ls2[wave][col] = s2[v];
            if (wave == (i / CHUNK)) lpos[col] = pos[v];   // diag tile's chunk
        }
    }
    __syncthreads();

    // First 16 threads: merge 4 chunk states per row, emit loss terms.
    float local = 0.f;
    if (tid < 16) {
        int r = tid;
        float fm1 = lm1[0][r], fs1 = ls1[0][r];
        float fm2 = lm2[0][r], fs2 = ls2[0][r];
#pragma unroll
        for (int w = 1; w < WAVES; ++w) {
            lse_merge(fm1, fs1, lm1[w][r], ls1[w][r]);
            lse_merge(fm2, fs2, lm2[w][r], ls2[w][r]);
        }
        float lse1 = fm1 + __logf(fs1);     // logaddexp(lse11, lse12_row)
        float lse2 = fm2 + __logf(fs2);     // logaddexp(lse12_col, lse22)
        float p = lpos[r];
        local = ((lse1 - p) + (lse2 - p)) * (0.5f / (float)(NN * BB));
    }

    // Reduce within wave 0, one atomic per block.
    if (wave == 0) {
#pragma unroll
        for (int mask = 16; mask >= 1; mask >>= 1)
            local += __shfl_xor(local, mask, 32);
        if (lane == 0) atomicAdd(out, local);
    }
}

__global__ void zero_out_kernel(float* out) { out[0] = 0.0f; }

extern "C" void kernel_launch(void* const* d_in, const int* in_sizes, int n_in,
                              void* d_out, int out_size, void* d_ws, size_t ws_size,
                              hipStream_t stream) {
    (void)in_sizes; (void)n_in; (void)d_ws; (void)ws_size; (void)out_size;
    const float* X1 = (const float*)d_in[0];
    const float* X2 = (const float*)d_in[1];
    float* out = (float*)d_out;

    zero_out_kernel<<<1, 1, 0, stream>>>(out);
    const int tasks = BB * NTILE;   // 2048 blocks x 4 waves
    infonce_fused_kernel<<<tasks, 32 * WAVES, 0, stream>>>(X1, X2, out);
}